// HGTJudge_46471546143471
// MI455X (gfx1250) — compile-verified
//
#include <hip/hip_runtime.h>
#include <math.h>

typedef float v2f __attribute__((ext_vector_type(2)));
typedef float v8f __attribute__((ext_vector_type(8)));

#define C_DIM 128
#define H_DIM 8
#define D_DIM 16

__device__ __forceinline__ float gelu_exact(float x) {
  return 0.5f * x * (1.0f + erff(x * 0.70710678118654752f));
}

// float atomic max via sign-split integer atomics (init to -inf bits)
__device__ __forceinline__ void atomic_max_f32(float* addr, float val) {
  if (val >= 0.0f) atomicMax((int*)addr, __float_as_int(val));
  else             atomicMin((unsigned int*)addr, __float_as_uint(val));
}

__global__ void fill_f32_kernel(float* __restrict__ p, float v, int n) {
  int i = blockIdx.x * blockDim.x + threadIdx.x;
  if (i < n) p[i] = v;
}

// Elementwise exact GELU, float4-vectorized (n must be a multiple of 4; N*128 is).
__global__ __launch_bounds__(256)
void gelu_kernel(const float* __restrict__ in, float* __restrict__ out, int n4) {
  int i = blockIdx.x * blockDim.x + threadIdx.x;
  if (i >= n4) return;
  const float4 v = ((const float4*)in)[i];
  float4 o;
  o.x = gelu_exact(v.x);
  o.y = gelu_exact(v.y);
  o.z = gelu_exact(v.z);
  o.w = gelu_exact(v.w);
  ((float4*)out)[i] = o;
}

// ---------------------------------------------------------------------------
// Dense GEMM: out[N,128] = epi( A[N,KDIM] @ W[KDIM,128] + bias )
// epi: 0 = none, 1 = relu, 2 = beta*o + (1-beta)*prev  (beta = sigmoid(skip))
// One wave computes a 16x128 strip via 8 accumulators of v_wmma_f32_16x16x4_f32.
// W (<=64KB) staged in LDS once per block.
// ---------------------------------------------------------------------------
template<int KDIM>
__global__ __launch_bounds__(256)
void gemm_wmma_kernel(const float* __restrict__ A, const float* __restrict__ W,
                      const float* __restrict__ bias, float* __restrict__ out,
                      const float* __restrict__ prev, const float* __restrict__ skipv,
                      int N, int epi)
{
  __shared__ float sW[KDIM * C_DIM];
  for (int i = threadIdx.x; i < KDIM * C_DIM; i += 256) sW[i] = W[i];
  __syncthreads();

  const int wave = threadIdx.x >> 5;
  const int lane = threadIdx.x & 31;
  const int half = lane >> 4;      // 0: K pair 0/1, 1: K pair 2/3
  const int m    = lane & 15;      // M for A-frag, N for B/C/D-frag
  const int row0 = blockIdx.x * 128 + wave * 16;

  int arow = row0 + m;
  if (arow >= N) arow = N - 1;     // clamped load; stores are guarded
  const float* Ar = A + (size_t)arow * KDIM;

  v8f acc[8];
#pragma unroll
  for (int i = 0; i < 8; ++i)
#pragma unroll
    for (int j = 0; j < 8; ++j) acc[i][j] = 0.0f;

#pragma unroll 4
  for (int s = 0; s < KDIM / 4; ++s) {
    const int kk = s * 4 + half * 2;
    const float2 a2 = *(const float2*)&Ar[kk];
    v2f af; af.x = a2.x; af.y = a2.y;
#pragma unroll
    for (int nt = 0; nt < 8; ++nt) {
      const int col = nt * 16 + m;
      v2f bf; bf.x = sW[kk * C_DIM + col]; bf.y = sW[(kk + 1) * C_DIM + col];
      acc[nt] = __builtin_amdgcn_wmma_f32_16x16x4_f32(
          false, af, false, bf, (short)0, acc[nt], false, false);
    }
  }

  float beta = 0.0f, omb = 0.0f;
  if (epi == 2) {
    float sv = skipv[0];
    beta = 1.0f / (1.0f + expf(-sv));
    omb  = 1.0f - beta;
  }

#pragma unroll
  for (int nt = 0; nt < 8; ++nt) {
    const int col = nt * 16 + m;
    const float bvv = bias[col];
#pragma unroll
    for (int r = 0; r < 8; ++r) {
      const int orow = row0 + r + 8 * half;   // C/D layout: VGPR r -> M=r (lo half) / r+8 (hi half)
      if (orow < N) {
        float o = acc[nt][r] + bvv;
        if (epi == 1)      o = fmaxf(o, 0.0f);
        else if (epi == 2) o = beta * o + omb * prev[(size_t)orow * C_DIM + col];
        out[(size_t)orow * C_DIM + col] = o;
      }
    }
  }
}

// ---------------------------------------------------------------------------
// Edge pass 1: alpha[e,h] = (q[dst,h,:] . (k[src,h,:] @ a_rel[h])) * p[h] / sqrt(D)
// plus segment max via float atomic max. One thread per (edge, head).
// ---------------------------------------------------------------------------
__global__ __launch_bounds__(256)
void edge_alpha_kernel(const float* __restrict__ k_s, const float* __restrict__ q_d,
                       const int* __restrict__ edge, const float* __restrict__ a_rel,
                       const float* __restrict__ p_rel, float* __restrict__ alpha,
                       float* __restrict__ amax, int E)
{
  __shared__ float sA[H_DIM * D_DIM * D_DIM];
  __shared__ float sP[H_DIM];
  for (int i = threadIdx.x; i < H_DIM * D_DIM * D_DIM; i += blockDim.x) sA[i] = a_rel[i];
  if (threadIdx.x < H_DIM) sP[threadIdx.x] = p_rel[threadIdx.x];
  __syncthreads();

  int idx = blockIdx.x * blockDim.x + threadIdx.x;
  if (idx >= E * H_DIM) return;
  const int e = idx >> 3, h = idx & 7;
  const int src = edge[e], dst = edge[E + e];

  const float4* kp = (const float4*)(k_s + (size_t)src * C_DIM + h * D_DIM);
  const float4* qp = (const float4*)(q_d + (size_t)dst * C_DIM + h * D_DIM);
  float kv[D_DIM], qv[D_DIM];
#pragma unroll
  for (int i = 0; i < 4; ++i) {
    const float4 k4 = kp[i], q4 = qp[i];
    kv[4*i+0] = k4.x; kv[4*i+1] = k4.y; kv[4*i+2] = k4.z; kv[4*i+3] = k4.w;
    qv[4*i+0] = q4.x; qv[4*i+1] = q4.y; qv[4*i+2] = q4.z; qv[4*i+3] = q4.w;
  }

  const float* Ah = &sA[h * D_DIM * D_DIM];
  float al = 0.0f;
#pragma unroll
  for (int f = 0; f < D_DIM; ++f) {
    float t = 0.0f;
#pragma unroll
    for (int d = 0; d < D_DIM; ++d) t = fmaf(kv[d], Ah[d * D_DIM + f], t);
    al = fmaf(qv[f], t, al);
  }
  al *= sP[h] * 0.25f;   // 1/sqrt(16)
  alpha[idx] = al;
  atomic_max_f32(&amax[dst * H_DIM + h], al);
}

// Edge pass 2: exp(alpha - max) in place + segment sum of denominators.
__global__ __launch_bounds__(256)
void edge_exp_kernel(const int* __restrict__ edge, float* __restrict__ alpha,
                     const float* __restrict__ amax, float* __restrict__ denom, int E)
{
  int idx = blockIdx.x * blockDim.x + threadIdx.x;
  if (idx >= E * H_DIM) return;
  const int e = idx >> 3, h = idx & 7;
  const int dst = edge[E + e];
  const float ex = expf(alpha[idx] - amax[dst * H_DIM + h]);
  alpha[idx] = ex;
  atomicAdd(&denom[dst * H_DIM + h], ex);
}

// Edge pass 3: agg[dst,h,:] += w * (v[src,h,:] @ m_rel[h])
__global__ __launch_bounds__(256)
void edge_agg_kernel(const float* __restrict__ v_s, const int* __restrict__ edge,
                     const float* __restrict__ m_rel, const float* __restrict__ alphaE,
                     const float* __restrict__ denom, float* __restrict__ agg, int E)
{
  __shared__ float sM[H_DIM * D_DIM * D_DIM];
  for (int i = threadIdx.x; i < H_DIM * D_DIM * D_DIM; i += blockDim.x) sM[i] = m_rel[i];
  __syncthreads();

  int idx = blockIdx.x * blockDim.x + threadIdx.x;
  if (idx >= E * H_DIM) return;
  const int e = idx >> 3, h = idx & 7;
  const int src = edge[e], dst = edge[E + e];

  const float w = alphaE[idx] / fmaxf(denom[dst * H_DIM + h], 1e-16f);
  const float4* vp = (const float4*)(v_s + (size_t)src * C_DIM + h * D_DIM);
  float vv[D_DIM];
#pragma unroll
  for (int i = 0; i < 4; ++i) {
    const float4 v4 = vp[i];
    vv[4*i+0] = v4.x; vv[4*i+1] = v4.y; vv[4*i+2] = v4.z; vv[4*i+3] = v4.w;
  }

  const float* Mh = &sM[h * D_DIM * D_DIM];
  float* ag = agg + (size_t)dst * C_DIM + h * D_DIM;
#pragma unroll
  for (int f = 0; f < D_DIM; ++f) {
    float t = 0.0f;
#pragma unroll
    for (int d = 0; d < D_DIM; ++d) t = fmaf(vv[d], Mh[d * D_DIM + f], t);
    atomicAdd(&ag[f], w * t);
  }
}

// Column-sum of h0 into g[128] (atomic partials)
__global__ __launch_bounds__(128)
void colsum_kernel(const float* __restrict__ h0, float* __restrict__ g, int N)
{
  const int col = threadIdx.x;
  float s = 0.0f;
  for (int r = blockIdx.x; r < N; r += gridDim.x) s += h0[(size_t)r * C_DIM + col];
  atomicAdd(&g[col], s);
}

// Final classifier: mean, relu(g@Wc1+bc1) @ Wc2 + bc2 -> sigmoid
__global__ __launch_bounds__(128)
void classifier_kernel(const float* __restrict__ g, const float* __restrict__ Wc1,
                       const float* __restrict__ bc1, const float* __restrict__ Wc2,
                       const float* __restrict__ bc2, float* __restrict__ out, int N)
{
  __shared__ float sg[C_DIM];
  __shared__ float red[C_DIM];
  const int t = threadIdx.x;
  sg[t] = g[t] / (float)N;
  __syncthreads();
  float z = bc1[t];
  for (int c = 0; c < C_DIM; ++c) z = fmaf(sg[c], Wc1[c * C_DIM + t], z);
  z = fmaxf(z, 0.0f);
  red[t] = z * Wc2[t];
  __syncthreads();
  for (int s = 64; s > 0; s >>= 1) {
    if (t < s) red[t] += red[t + s];
    __syncthreads();
  }
  if (t == 0) out[0] = 1.0f / (1.0f + expf(-(red[0] + bc2[0])));
}

// ---------------------------------------------------------------------------
extern "C" void kernel_launch(void* const* d_in, const int* in_sizes, int n_in,
                              void* d_out, int out_size, void* d_ws, size_t ws_size,
                              hipStream_t stream)
{
  const float* x_arg   = (const float*)d_in[0];
  const float* x_ev    = (const float*)d_in[1];
  const int*   edge_aa = (const int*)d_in[2];
  const int*   edge_ea = (const int*)d_in[3];
  const int*   edge_ae = (const int*)d_in[4];
  const float* Win_arg = (const float*)d_in[5];
  const float* bin_arg = (const float*)d_in[6];
  const float* Win_ev  = (const float*)d_in[7];
  const float* bin_ev  = (const float*)d_in[8];
  const float* Wk = (const float*)d_in[9];
  const float* bk = (const float*)d_in[10];
  const float* Wq = (const float*)d_in[11];
  const float* bq = (const float*)d_in[12];
  const float* Wv = (const float*)d_in[13];
  const float* bvp = (const float*)d_in[14];
  const float* a_rel = (const float*)d_in[15];
  const float* m_rel = (const float*)d_in[16];
  const float* p_rel = (const float*)d_in[17];
  const float* Wa = (const float*)d_in[18];
  const float* ba = (const float*)d_in[19];
  const float* skip = (const float*)d_in[20];
  const float* Wc1 = (const float*)d_in[21];
  const float* bc1 = (const float*)d_in[22];
  const float* Wc2 = (const float*)d_in[23];
  const float* bc2 = (const float*)d_in[24];

  const int F_IN = 64;
  const int N0 = in_sizes[0] / F_IN;
  const int N1 = in_sizes[1] / F_IN;
  const int E  = in_sizes[2] / 2;
  const int L  = 2;
  const int Nmax = (N0 > N1) ? N0 : N1;

  // ---- workspace carve ----
  char* wsb = (char*)d_ws;
  size_t off = 0;
  auto carve = [&](size_t bytes) -> float* {
    float* p = (float*)(wsb + off);
    off = (off + bytes + 255) & ~(size_t)255;
    return p;
  };
  float* h[2];   h[0]  = carve((size_t)N0 * C_DIM * 4); h[1]  = carve((size_t)N1 * C_DIM * 4);
  float* kk[2];  kk[0] = carve((size_t)N0 * C_DIM * 4); kk[1] = carve((size_t)N1 * C_DIM * 4);
  float* qq[2];  qq[0] = carve((size_t)N0 * C_DIM * 4); qq[1] = carve((size_t)N1 * C_DIM * 4);
  float* vv[2];  vv[0] = carve((size_t)N0 * C_DIM * 4); vv[1] = carve((size_t)N1 * C_DIM * 4);
  float* agg[2]; agg[0] = carve((size_t)N0 * C_DIM * 4); agg[1] = carve((size_t)N1 * C_DIM * 4);
  float* alpha = carve((size_t)E * H_DIM * 4);
  float* amax  = carve((size_t)Nmax * H_DIM * 4);
  float* denom = carve((size_t)Nmax * H_DIM * 4);
  float* gbuf  = carve(C_DIM * 4);

  const dim3 blk256(256), blk128(128);

  // ---- input projections (K=64, relu epilogue) ----
  gemm_wmma_kernel<64><<<dim3((N0 + 127) / 128), blk256, 0, stream>>>(
      x_arg, Win_arg, bin_arg, h[0], nullptr, nullptr, N0, 1);
  gemm_wmma_kernel<64><<<dim3((N1 + 127) / 128), blk256, 0, stream>>>(
      x_ev, Win_ev, bin_ev, h[1], nullptr, nullptr, N1, 1);

  const int rel_s[3] = {0, 1, 0};
  const int rel_d[3] = {0, 0, 1};
  const int* rel_e[3] = {edge_aa, edge_ea, edge_ae};

  for (int l = 0; l < L; ++l) {
    // K/Q/V projections per node type
    for (int t = 0; t < 2; ++t) {
      const int N = t ? N1 : N0;
      const dim3 grd((N + 127) / 128);
      const size_t wOff = ((size_t)l * 2 + t) * C_DIM * C_DIM;
      const size_t bOff = ((size_t)l * 2 + t) * C_DIM;
      gemm_wmma_kernel<128><<<grd, blk256, 0, stream>>>(h[t], Wk + wOff, bk + bOff, kk[t], nullptr, nullptr, N, 0);
      gemm_wmma_kernel<128><<<grd, blk256, 0, stream>>>(h[t], Wq + wOff, bq + bOff, qq[t], nullptr, nullptr, N, 0);
      gemm_wmma_kernel<128><<<grd, blk256, 0, stream>>>(h[t], Wv + wOff, bvp + bOff, vv[t], nullptr, nullptr, N, 0);
    }
    hipMemsetAsync(agg[0], 0, (size_t)N0 * C_DIM * 4, stream);
    hipMemsetAsync(agg[1], 0, (size_t)N1 * C_DIM * 4, stream);

    for (int r = 0; r < 3; ++r) {
      const int sT = rel_s[r], dT = rel_d[r];
      const int Nd = dT ? N1 : N0;
      const float* ar = a_rel + ((size_t)l * 3 + r) * H_DIM * D_DIM * D_DIM;
      const float* mr = m_rel + ((size_t)l * 3 + r) * H_DIM * D_DIM * D_DIM;
      const float* pr = p_rel + ((size_t)l * 3 + r) * H_DIM;

      const int nF = Nd * H_DIM;
      fill_f32_kernel<<<dim3((nF + 255) / 256), blk256, 0, stream>>>(amax, -INFINITY, nF);
      hipMemsetAsync(denom, 0, (size_t)nF * 4, stream);

      const int nEH = E * H_DIM;
      const dim3 gE((nEH + 255) / 256);
      edge_alpha_kernel<<<gE, blk256, 0, stream>>>(kk[sT], qq[dT], rel_e[r], ar, pr, alpha, amax, E);
      edge_exp_kernel<<<gE, blk256, 0, stream>>>(rel_e[r], alpha, amax, denom, E);
      edge_agg_kernel<<<gE, blk256, 0, stream>>>(vv[sT], rel_e[r], mr, alpha, denom, agg[dT], E);
    }

    // output projection: standalone GELU pass (into dead kk buffer), then
    // branch-free WMMA GEMM with sigmoid-skip blend (in-place on h)
    for (int t = 0; t < 2; ++t) {
      const int N = t ? N1 : N0;
      const int n4 = N * C_DIM / 4;
      gelu_kernel<<<dim3((n4 + 255) / 256), blk256, 0, stream>>>(agg[t], kk[t], n4);
      const dim3 grd((N + 127) / 128);
      const size_t wOff = ((size_t)l * 2 + t) * C_DIM * C_DIM;
      const size_t bOff = ((size_t)l * 2 + t) * C_DIM;
      gemm_wmma_kernel<128><<<grd, blk256, 0, stream>>>(
          kk[t], Wa + wOff, ba + bOff, h[t], h[t], skip + (size_t)l * 2 + t, N, 2);
    }
  }

  // ---- readout ----
  hipMemsetAsync(gbuf, 0, C_DIM * 4, stream);
  colsum_kernel<<<dim3(64), blk128, 0, stream>>>(h[0], gbuf, N0);
  classifier_kernel<<<dim3(1), blk128, 0, stream>>>(gbuf, Wc1, bc1, Wc2, bc2, (float*)d_out, N0);
}